// AttentionLayer_51642686767167
// MI455X (gfx1250) — compile-verified
//
#include <hip/hip_runtime.h>
#include <math.h>

// ---------------------------------------------------------------------------
// B=64, T=512, D=1024
//   q[b,:]   = W @ x[b,511,:]                    (K1: WMMA f32 16x16x4 GEMM)
//   s[b,t]   = x[b,t,:] . q[b]   t in [0,511)    (K2: stream x, b128 loads)
//   alpha    = softmax_t(s)                      (K3: tiny per-b kernel)
//   c[b,d]   = sum_t alpha[b,t] * x[b,t,d]       (K4: partials, K5: reduce)
//   out[b,:] = concat(c[b,:], x[b,511,:])        (64 x 2048)
// Roofline: ~268 MFLOP vs 134 MB streamed (x2 passes, 2nd is L2-resident:
// 134 MB < 192 MB L2). HBM floor ~6 us at 23.3 TB/s -> bandwidth bound, so
// the streaming passes use 512 blocks (b x 8 t-chunks) and float4/b128 loads.
// ---------------------------------------------------------------------------

typedef __attribute__((ext_vector_type(2))) float v2f;
typedef __attribute__((ext_vector_type(4))) float v4f;
typedef __attribute__((ext_vector_type(8))) float v8f;

#define BATCH  64
#define TDIM   512
#define DDIM   1024
#define THIST  511   // T-1
#define TCHUNK 64    // t rows per streaming block
#define NCHUNK 8     // ceil(511/64)

// ---------------------------------------------------------------------------
// K1: Q = X_last (64x1024) @ W^T (1024x1024) via V_WMMA_F32_16X16X4_F32.
// One wave per 16x16 tile: 4 x 64 = 256 waves = 32 blocks x 8 waves.
// A(16x4): lanes 0-15 hold M=lane {K=klo,klo+1}, lanes 16-31 klo=2.
// B(4x16): lane holds B[K=klo+{0,1}, N=lane&15] = W[n0+(lane&15), k0+klo+{0,1}].
// D(16x16): VGPR r -> row M = r + 8*(lane>=16), col N = lane&15.
// ---------------------------------------------------------------------------
__global__ __launch_bounds__(256)
void qgemm_wmma_kernel(const float* __restrict__ x,
                       const float* __restrict__ W,
                       float* __restrict__ qbuf)
{
    const int lane  = threadIdx.x & 31;
    const int gwave = blockIdx.x * 8 + (threadIdx.x >> 5);   // 0..255
    const int m0 = (gwave >> 6) * 16;
    const int n0 = (gwave & 63) * 16;

    const int half = lane >> 4;
    const int l16  = lane & 15;
    const int klo  = half * 2;

    const float* __restrict__ arow =
        x + ((size_t)(m0 + l16) * TDIM + (TDIM - 1)) * DDIM;   // x_last row
    const float* __restrict__ brow = W + (size_t)(n0 + l16) * DDIM;

    v8f acc = {};
#pragma unroll 8
    for (int k0 = 0; k0 < DDIM; k0 += 4) {
        v2f a = *reinterpret_cast<const v2f*>(arow + k0 + klo);
        v2f b = *reinterpret_cast<const v2f*>(brow + k0 + klo);
        acc = __builtin_amdgcn_wmma_f32_16x16x4_f32(
            false, a, false, b, (short)0, acc, false, false);
    }

#pragma unroll
    for (int r = 0; r < 8; ++r)
        qbuf[(size_t)(m0 + r + half * 8) * DDIM + n0 + l16] = acc[r];
}

// ---------------------------------------------------------------------------
// K2: scores[b,t] = x[b,t,:] . q[b].  Grid: 64 b x 8 chunks = 512 blocks,
// 256 threads (8 waves). Wave handles 8 rows; per row: 8 iters of float4
// (global_load_b128) + ds b128 reads of q, then a wave32 shuffle reduction.
// ---------------------------------------------------------------------------
__global__ __launch_bounds__(256)
void scores_kernel(const float* __restrict__ x,
                   const float* __restrict__ qbuf,
                   float* __restrict__ scores)
{
    __shared__ float q_s[DDIM];

    const int b     = blockIdx.x >> 3;
    const int chunk = blockIdx.x & 7;
    const int tid   = threadIdx.x;
    const int lane  = tid & 31;
    const int wave  = tid >> 5;

    *reinterpret_cast<v4f*>(q_s + tid * 4) =
        *reinterpret_cast<const v4f*>(qbuf + (size_t)b * DDIM + tid * 4);
    __syncthreads();

    const float* __restrict__ xb = x + (size_t)b * TDIM * DDIM;
    const int t0 = chunk * TCHUNK;

    for (int t = t0 + wave; t < t0 + TCHUNK && t < THIST; t += 8) {
        const float* __restrict__ xr = xb + (size_t)t * DDIM;
        if (t + 8 < THIST)
            __builtin_prefetch(xb + (size_t)(t + 8) * DDIM + lane * 4, 0, 1);
        v4f p4 = {};
#pragma unroll
        for (int i = 0; i < 8; ++i) {
            const int d = i * 128 + lane * 4;
            v4f xv = *reinterpret_cast<const v4f*>(xr + d);
            v4f qv = *reinterpret_cast<const v4f*>(q_s + d);
            p4[0] = fmaf(xv[0], qv[0], p4[0]);
            p4[1] = fmaf(xv[1], qv[1], p4[1]);
            p4[2] = fmaf(xv[2], qv[2], p4[2]);
            p4[3] = fmaf(xv[3], qv[3], p4[3]);
        }
        float partial = (p4[0] + p4[1]) + (p4[2] + p4[3]);
#pragma unroll
        for (int m = 16; m >= 1; m >>= 1)
            partial += __shfl_xor(partial, m, 32);
        if (lane == 0) scores[(size_t)b * 512 + t] = partial;
    }
}

// ---------------------------------------------------------------------------
// K3: softmax over 511 scores per b; writes normalized alpha in place
// (slot 511 gets 0). 64 blocks x 512 threads (16 waves).
// ---------------------------------------------------------------------------
__global__ __launch_bounds__(512)
void softmax_kernel(float* __restrict__ scores)
{
    __shared__ float red_s[16];
    __shared__ float stat_s[2];

    const int b    = blockIdx.x;
    const int tid  = threadIdx.x;
    const int lane = tid & 31;
    const int wave = tid >> 5;   // 0..15

    float* __restrict__ sb = scores + (size_t)b * 512;
    const float sv = (tid < THIST) ? sb[tid] : -INFINITY;

    float mx = sv;
#pragma unroll
    for (int k = 16; k >= 1; k >>= 1)
        mx = fmaxf(mx, __shfl_xor(mx, k, 32));
    if (lane == 0) red_s[wave] = mx;
    __syncthreads();
    if (wave == 0) {
        float v = (lane < 16) ? red_s[lane] : -INFINITY;
#pragma unroll
        for (int k = 16; k >= 1; k >>= 1)
            v = fmaxf(v, __shfl_xor(v, k, 32));
        if (lane == 0) stat_s[0] = v;
    }
    __syncthreads();

    const float e = (tid < THIST) ? __expf(sv - stat_s[0]) : 0.0f;
    float sum = e;
#pragma unroll
    for (int k = 16; k >= 1; k >>= 1)
        sum += __shfl_xor(sum, k, 32);
    if (lane == 0) red_s[wave] = sum;
    __syncthreads();
    if (wave == 0) {
        float v = (lane < 16) ? red_s[lane] : 0.0f;
#pragma unroll
        for (int k = 16; k >= 1; k >>= 1)
            v += __shfl_xor(v, k, 32);
        if (lane == 0) stat_s[1] = 1.0f / v;
    }
    __syncthreads();

    sb[tid] = e * stat_s[1];   // alpha; tid==511 -> 0
}

// ---------------------------------------------------------------------------
// K4: partial[b,chunk,d] = sum_{t in chunk} alpha[b,t] * x[b,t,d].
// Grid: 512 blocks x 256 threads; thread owns 4 consecutive d (b128 loads).
// Second pass over x is L2-resident (134 MB < 192 MB L2).
// ---------------------------------------------------------------------------
__global__ __launch_bounds__(256)
void wsum_partial_kernel(const float* __restrict__ x,
                         const float* __restrict__ alpha,
                         float* __restrict__ partial)
{
    __shared__ float a_s[TCHUNK];

    const int b     = blockIdx.x >> 3;
    const int chunk = blockIdx.x & 7;
    const int tid   = threadIdx.x;
    const int t0    = chunk * TCHUNK;

    if (tid < TCHUNK) {
        const int t = t0 + tid;
        a_s[tid] = (t < THIST) ? alpha[(size_t)b * 512 + t] : 0.0f;
    }
    __syncthreads();

    const float* __restrict__ xb = x + (size_t)b * TDIM * DDIM;
    const int d4 = tid * 4;
    const int tcnt = (t0 + TCHUNK <= THIST) ? TCHUNK : (THIST - t0);

    v4f acc = {};
    for (int i = 0; i < tcnt; ++i) {
        const float av = a_s[i];
        v4f xv = *reinterpret_cast<const v4f*>(xb + (size_t)(t0 + i) * DDIM + d4);
        acc[0] = fmaf(av, xv[0], acc[0]);
        acc[1] = fmaf(av, xv[1], acc[1]);
        acc[2] = fmaf(av, xv[2], acc[2]);
        acc[3] = fmaf(av, xv[3], acc[3]);
    }
    *reinterpret_cast<v4f*>(
        partial + ((size_t)b * NCHUNK + chunk) * DDIM + d4) = acc;
}

// ---------------------------------------------------------------------------
// K5: c[b,d] = sum_chunk partial; out = concat(c, x_last). 64 blocks x 1024.
// ---------------------------------------------------------------------------
__global__ __launch_bounds__(1024)
void finalize_kernel(const float* __restrict__ x,
                     const float* __restrict__ partial,
                     float* __restrict__ out)
{
    const int b = blockIdx.x;
    const int d = threadIdx.x;

    float acc = 0.0f;
#pragma unroll
    for (int c = 0; c < NCHUNK; ++c)
        acc += partial[((size_t)b * NCHUNK + c) * DDIM + d];

    out[(size_t)b * (2 * DDIM) + d]        = acc;
    out[(size_t)b * (2 * DDIM) + DDIM + d] =
        x[((size_t)b * TDIM + TDIM - 1) * DDIM + d];
}

extern "C" void kernel_launch(void* const* d_in, const int* in_sizes, int n_in,
                              void* d_out, int out_size, void* d_ws, size_t ws_size,
                              hipStream_t stream)
{
    const float* x = (const float*)d_in[0];   // (64, 512, 1024) f32
    const float* W = (const float*)d_in[1];   // (1024, 1024)    f32
    float* out = (float*)d_out;               // (64, 2048)      f32

    // Workspace layout (floats): qbuf[64*1024] | scores[64*512] | partial[64*8*1024]
    float* qbuf    = (float*)d_ws;
    float* scores  = qbuf + (size_t)BATCH * DDIM;
    float* partial = scores + (size_t)BATCH * 512;

    qgemm_wmma_kernel<<<32, 256, 0, stream>>>(x, W, qbuf);
    scores_kernel<<<BATCH * NCHUNK, 256, 0, stream>>>(x, qbuf, scores);
    softmax_kernel<<<BATCH, 512, 0, stream>>>(scores);
    wsum_partial_kernel<<<BATCH * NCHUNK, 256, 0, stream>>>(x, scores, partial);
    finalize_kernel<<<BATCH, 1024, 0, stream>>>(x, partial, out);
}